// SelfAttention_63720134804106
// MI455X (gfx1250) — compile-verified
//
#include <hip/hip_runtime.h>
#include <math.h>

#define N_HW 4096
#define BATCH 4

typedef float v2f __attribute__((ext_vector_type(2)));
typedef float v8f __attribute__((ext_vector_type(8)));

// D = A(16x4 f32) * B(4x16 f32) + C(16x16 f32), full f32 precision on the
// CDNA5 matrix pipe.
__device__ __forceinline__ v8f wmma_f32_16x16x4(v2f a, v2f b, v8f c) {
  return __builtin_amdgcn_wmma_f32_16x16x4_f32(
      /*neg_a=*/false, a, /*neg_b=*/false, b,
      /*c_mod=*/(short)0, c, /*reuse_a=*/false, /*reuse_b=*/false);
}

// ---------------------------------------------------------------------------
// Kernel 1: spectral norm of the 4 weight matrices, one wave per matrix.
//   v1 = l2n(w^T u); s = w v1; sigma = ||s||^2/(||s||+eps); w_sn = w/sigma
// ---------------------------------------------------------------------------
__global__ void sn_kernel(const float* __restrict__ wt, const float* __restrict__ wp,
                          const float* __restrict__ wg, const float* __restrict__ wo,
                          const float* __restrict__ ut, const float* __restrict__ up,
                          const float* __restrict__ ug, const float* __restrict__ uo,
                          float* __restrict__ ws) {
  __shared__ float t_sh[4][64];
  const int wv = threadIdx.x >> 5;
  const int ln = threadIdx.x & 31;
  const int odA[4] = {8, 8, 32, 64};
  const int idA[4] = {64, 64, 64, 32};
  const float* wA[4] = {wt, wp, wg, wo};
  const float* uA[4] = {ut, up, ug, uo};
  float* dA[4] = {ws, ws + 512, ws + 1024, ws + 3072};
  const int od = odA[wv], id = idA[wv];
  const float* w = wA[wv];
  const float* u = uA[wv];
  float* dst = dA[wv];

  // t = w^T u, lane-parallel over columns
  float ssq = 0.f;
  for (int c = ln; c < id; c += 32) {
    float acc = 0.f;
    for (int o = 0; o < od; ++o) acc += w[o * id + c] * u[o];
    t_sh[wv][c] = acc;
    ssq += acc * acc;
  }
  for (int off = 16; off > 0; off >>= 1) ssq += __shfl_xor(ssq, off);
  const float invn = 1.0f / (sqrtf(ssq) + 1e-12f);
  __syncthreads();  // uniform across all 4 waves

  // s = w * (t/||t||), sigma = ||s||^2 / (||s|| + eps)
  float ssq2 = 0.f;
  for (int o = ln; o < od; o += 32) {
    float acc = 0.f;
    for (int c = 0; c < id; ++c) acc += w[o * id + c] * t_sh[wv][c];
    acc *= invn;
    ssq2 += acc * acc;
  }
  for (int off = 16; off > 0; off >>= 1) ssq2 += __shfl_xor(ssq2, off);
  const float sigma = ssq2 / (sqrtf(ssq2) + 1e-12f);
  const float inv = 1.0f / sigma;
  for (int k = ln; k < od * id; k += 32) dst[k] = w[k] * inv;
}

// ---------------------------------------------------------------------------
// Kernel 2a: theta/phi projections in K-pair-interleaved layout [B][4][N][2]
// (channel pairs innermost -> flash kernel loads one b64 per WMMA operand).
// Each thread computes channel rows 2p and 2p+1 at column n, stores a float2.
// ---------------------------------------------------------------------------
__global__ void proj_tp_kernel(const float* __restrict__ x, const float* __restrict__ wt,
                               const float* __restrict__ wp,
                               v2f* __restrict__ theta_p, v2f* __restrict__ phi_p) {
  const int idx = blockIdx.x * blockDim.x + threadIdx.x;  // B*8*N threads
  const int n = idx & (N_HW - 1);
  const int p = (idx >> 12) & 3;
  const int s = (idx >> 14) & 1;  // 0 = theta, 1 = phi
  const int b = idx >> 15;
  const float* w = (s ? wp : wt) + (2 * p) * 64;
  const float* xb = x + b * 64 * N_HW + n;
  float acc0 = 0.f, acc1 = 0.f;
#pragma unroll 8
  for (int c = 0; c < 64; ++c) {
    const float xv = xb[c * N_HW];
    acc0 += w[c] * xv;
    acc1 += w[64 + c] * xv;
  }
  v2f r;
  r.x = acc0;
  r.y = acc1;
  (s ? phi_p : theta_p)[(b * 4 + p) * N_HW + n] = r;
}

// ---------------------------------------------------------------------------
// Kernel 2b: g projection in key-pair-interleaved layout [B][N/2][32][2]
// (key pairs innermost -> PV A-operand is one aligned b64 per lane).
// ---------------------------------------------------------------------------
__global__ void proj_g_kernel(const float* __restrict__ x, const float* __restrict__ wg,
                              v2f* __restrict__ g_p) {
  const int idx = blockIdx.x * blockDim.x + threadIdx.x;  // B*(N/2)*32 threads
  const int c32 = idx & 31;
  const int n2 = (idx >> 5) & (N_HW / 2 - 1);
  const int b = idx >> 16;
  const float* xb = x + b * 64 * N_HW + 2 * n2;
  const float* w = wg + c32 * 64;
  float acc0 = 0.f, acc1 = 0.f;
#pragma unroll 8
  for (int c = 0; c < 64; ++c) {
    const float wv = w[c];
    acc0 += wv * xb[c * N_HW];
    acc1 += wv * xb[c * N_HW + 1];
  }
  v2f r;
  r.x = acc0;
  r.y = acc1;
  g_p[(b * (N_HW / 2) + n2) * 32 + c32] = r;
}

// ---------------------------------------------------------------------------
// Kernel 3: fused flash attention. One wave owns a 16-query tile (queries=phi
// columns j), streams 256 key tiles (keys=theta columns i), online column
// softmax, PV accumulate with V=g (d_v=32). All matmuls via v_wmma_f32_16x16x4.
// All A/B operands are single b64 loads thanks to pair-interleaved layouts.
// ---------------------------------------------------------------------------
__global__ void __launch_bounds__(256) flash_kernel(const v2f* __restrict__ theta_p,
                                                    const v2f* __restrict__ phi_p,
                                                    const v2f* __restrict__ g_p,
                                                    float* __restrict__ o_ws) {
  const int lane = threadIdx.x & 31;
  const int wv = threadIdx.x >> 5;
  const int task = blockIdx.x * 8 + wv;  // 1024 tasks total
  const int b = task >> 8;               // 256 j-tiles per batch
  const int j0 = (task & 255) << 4;
  const int half = lane >> 4;  // ISA: K/M split across lane halves
  const int l16 = lane & 15;

  const v2f* th = theta_p + b * 4 * N_HW;
  const v2f* ph = phi_p + b * 4 * N_HW;
  const v2f* gt = g_p + b * (N_HW / 2) * 32;

  // B-tile (phi, K=c x N=j) is loop-invariant: pair index = half (K=0..3) / 2+half (K=4..7)
  const v2f bq0 = ph[(0 + half) * N_HW + j0 + l16];
  const v2f bq1 = ph[(2 + half) * N_HW + j0 + l16];

  v8f O0 = {};  // c = 0..15  (rows r / r+8 in lane halves)
  v8f O1 = {};  // c = 16..31
  float m = -INFINITY, l = 0.f;

  for (int it = 0; it < 256; ++it) {
    const int i0 = it << 4;
    // A-tile (theta^T, M=i x K=c): one b64 per operand, coalesced over i
    const v2f a0 = th[(0 + half) * N_HW + i0 + l16];
    const v2f a1 = th[(2 + half) * N_HW + i0 + l16];
    v8f S = {};
    S = wmma_f32_16x16x4(a0, bq0, S);  // K = 0..3
    S = wmma_f32_16x16x4(a1, bq1, S);  // K = 4..7

    // online softmax over rows i for each column j (replicated across halves)
    float tm = S[0];
#pragma unroll
    for (int r = 1; r < 8; ++r) tm = fmaxf(tm, S[r]);
    tm = fmaxf(tm, __shfl_xor(tm, 16));
    const float mn = fmaxf(m, tm);
    const float sc = __expf(m - mn);  // first iter: exp(-inf)=0
    float P[8];
    float ts = 0.f;
#pragma unroll
    for (int r = 0; r < 8; ++r) {
      P[r] = __expf(S[r] - mn);
      ts += P[r];
    }
    ts += __shfl_xor(ts, 16);
    l = l * sc + ts;
    m = mn;
#pragma unroll
    for (int r = 0; r < 8; ++r) {
      O0[r] *= sc;
      O1[r] *= sc;
    }

    // PV: O(32xj) += g(32 x 16i) * P(16i x 16j), K in chunks of 4
#pragma unroll
    for (int kc = 0; kc < 16; kc += 4) {
      // re-lay P from D-layout to B-layout with half-crossing shuffles
      v2f bp;
      if (kc < 8) {
        const float tx = __shfl_xor(P[kc + 2], 16);
        const float ty = __shfl_xor(P[kc + 3], 16);
        bp.x = half ? tx : P[kc];
        bp.y = half ? ty : P[kc + 1];
      } else {
        const int r = kc - 8;
        const float tx = __shfl_xor(P[r], 16);
        const float ty = __shfl_xor(P[r + 1], 16);
        bp.x = half ? P[r + 2] : tx;
        bp.y = half ? P[r + 3] : ty;
      }
      // A-tile from g: key pair (row, row+1), row = i0+kc+2*half (even)
      const int pr = (i0 + kc) / 2 + half;  // key-pair index
      const v2f ag0 = gt[pr * 32 + l16];
      const v2f ag1 = gt[pr * 32 + 16 + l16];
      O0 = wmma_f32_16x16x4(ag0, bp, O0);
      O1 = wmma_f32_16x16x4(ag1, bp, O1);
    }
  }

  const float inv = 1.0f / l;
#pragma unroll
  for (int r = 0; r < 8; ++r) {
    O0[r] *= inv;
    O1[r] *= inv;
  }
  // o stored [B][32][N]; D-layout row r maps to c = r + 8*half
#pragma unroll
  for (int r = 0; r < 8; ++r) {
    o_ws[(b * 32 + (r + 8 * half)) * N_HW + j0 + l16] = O0[r];
    o_ws[(b * 32 + 16 + (r + 8 * half)) * N_HW + j0 + l16] = O1[r];
  }
}

// ---------------------------------------------------------------------------
// Kernel 4: out = gamma * (w_o @ o) + x
// ---------------------------------------------------------------------------
__global__ void out_kernel(const float* __restrict__ o_ws, const float* __restrict__ wo,
                           const float* __restrict__ x, const float* __restrict__ gamma,
                           float* __restrict__ out) {
  const int idx = blockIdx.x * blockDim.x + threadIdx.x;
  const int n = idx & (N_HW - 1);
  const int oc = (idx >> 12) & 63;
  const int b = idx >> 18;
  const float* ob = o_ws + b * 32 * N_HW + n;
  float acc = 0.f;
#pragma unroll 8
  for (int c = 0; c < 32; ++c) acc += wo[oc * 32 + c] * ob[c * N_HW];
  out[idx] = gamma[0] * acc + x[idx];
}

// ---------------------------------------------------------------------------
extern "C" void kernel_launch(void* const* d_in, const int* in_sizes, int n_in,
                              void* d_out, int out_size, void* d_ws, size_t ws_size,
                              hipStream_t stream) {
  const float* x = (const float*)d_in[0];
  const float* wt = (const float*)d_in[1];
  const float* wp = (const float*)d_in[2];
  const float* wg = (const float*)d_in[3];
  const float* wo = (const float*)d_in[4];
  const float* gamma = (const float*)d_in[5];
  const float* ut = (const float*)d_in[6];
  const float* up = (const float*)d_in[8];
  const float* ug = (const float*)d_in[10];
  const float* uo = (const float*)d_in[12];
  // (v_* inputs are unused by the reference's power-iteration step)

  float* ws = (float*)d_ws;
  float* wt_sn = ws;                           //  8x64
  float* wp_sn = ws + 512;                     //  8x64
  float* wg_sn = ws + 1024;                    // 32x64
  float* wo_sn = ws + 3072;                    // 64x32
  float* theta = ws + 5120;                    // [B][4][N][2] pair-interleaved
  float* phi = theta + BATCH * 8 * N_HW;       // [B][4][N][2]
  float* g_t = phi + BATCH * 8 * N_HW;         // [B][N/2][32][2]
  float* o_ws = g_t + BATCH * N_HW * 32;       // [B][32][N]
  float* out = (float*)d_out;

  sn_kernel<<<1, 128, 0, stream>>>(wt, wp, wg, wo, ut, up, ug, uo, ws);
  proj_tp_kernel<<<(BATCH * 8 * N_HW) / 256, 256, 0, stream>>>(
      x, wt_sn, wp_sn, (v2f*)theta, (v2f*)phi);
  proj_g_kernel<<<(BATCH * (N_HW / 2) * 32) / 256, 256, 0, stream>>>(x, wg_sn, (v2f*)g_t);
  flash_kernel<<<128, 256, 0, stream>>>((const v2f*)theta, (const v2f*)phi, (const v2f*)g_t,
                                        o_ws);
  out_kernel<<<(BATCH * 64 * N_HW) / 256, 256, 0, stream>>>(o_ws, wo_sn, x, gamma, out);
}